// DMLoss_50723563765992
// MI455X (gfx1250) — compile-verified
//
#include <hip/hip_runtime.h>
#include <hip/hip_bf16.h>
#include <math.h>

typedef __attribute__((ext_vector_type(2))) float v2f;
typedef __attribute__((ext_vector_type(8))) float v8f;

#define NP   128          // pred points per batch
#define NG   128          // gt vertices per batch
#define TME  10           // interpolation steps
#define NI   (NG * TME)   // 1280 interpolated gt points

// D = A(16x4) * B(4x16) + C, f32. With A-row=[gx,gy,|g|^2,1], B-col=[-2px,-2py,1,|p|^2]
// D[m][n] = squared distance between g_m and p_n, exact.
__device__ __forceinline__ v8f wmma_dist(v2f a, v2f b) {
    v8f z = {};
    return __builtin_amdgcn_wmma_f32_16x16x4_f32(
        /*neg_a=*/false, a, /*neg_b=*/false, b,
        /*c_mod=*/(short)0, z, /*reuse_a=*/false, /*reuse_b=*/false);
}

__launch_bounds__(128)
__global__ void dmloss_batch_kernel(const float* __restrict__ ini_pred,   // [B,NP,2]
                                    const float* __restrict__ pred2,      // [B,NP,2]
                                    const float* __restrict__ gt,         // [B,NG,2]
                                    const float* __restrict__ kmask,      // [B,NG]
                                    float* __restrict__ ws)               // [3] global accum
{
    __shared__ float sIx[NI];     // interpolated gt x
    __shared__ float sIy[NI];     // interpolated gt y
    __shared__ float sPx[NP];     // ini_pred x
    __shared__ float sPy[NP];     // ini_pred y
    __shared__ float sGx[NG];     // gt x
    __shared__ float sGy[NG];     // gt y
    __shared__ float sAcc[3];     // {pred2gt_sum, gt2pred_masked_sum, mask_sum}

    const int b    = blockIdx.x;
    const int tid  = threadIdx.x;        // 0..127
    const int lane = tid & 31;
    const int wave = tid >> 5;           // 0..3
    const int n    = lane & 15;          // row/col index within 16-tile
    const bool hi  = (lane >= 16);

    if (tid < 3) sAcc[tid] = 0.0f;

    // ---- stage 0: load points, build interpolation in LDS ----
    {
        const int g = tid;  // 0..127 (NP == NG == blockDim)
        sGx[g] = gt[(b * NG + g) * 2 + 0];
        sGy[g] = gt[(b * NG + g) * 2 + 1];
        sPx[g] = ini_pred[(b * NP + g) * 2 + 0];
        sPy[g] = ini_pred[(b * NP + g) * 2 + 1];
    }
    __syncthreads();
    {
        const int g = tid;
        const int p = (g + NG - 1) & (NG - 1);   // roll(shift=1): previous vertex
        const float cx = sGx[g], cy = sGy[g];
        const float qx = sGx[p], qy = sGy[p];
#pragma unroll
        for (int t = 0; t < TME; ++t) {
            const float w = (float)t / (float)TME;
            sIx[g * TME + t] = cx * w + qx * (1.0f - w);
            sIy[g * TME + t] = cy * w + qy * (1.0f - w);
        }
    }
    __syncthreads();

    const int c0 = wave * 2;      // this wave's two 16-column tiles
    const int c1 = wave * 2 + 1;

    // ================= stage 1: pred -> nearest interpolated gt =================
    // rows (M) = interp points, cols (N) = ini_pred points; argmin over rows.
    v2f b0, b1;
    {
        float px = sPx[c0 * 16 + n], py = sPy[c0 * 16 + n];
        b0[0] = hi ? 1.0f : -2.0f * px;
        b0[1] = hi ? (px * px + py * py) : -2.0f * py;
        px = sPx[c1 * 16 + n]; py = sPy[c1 * 16 + n];
        b1[0] = hi ? 1.0f : -2.0f * px;
        b1[1] = hi ? (px * px + py * py) : -2.0f * py;
    }
    float min0 = 3.0e38f, min1 = 3.0e38f;
    int   idx0 = 0,       idx1 = 0;
    for (int t = 0; t < NI / 16; ++t) {
        const float gx = sIx[t * 16 + n];
        const float gy = sIy[t * 16 + n];
        v2f a;
        a[0] = hi ? (gx * gx + gy * gy) : gx;
        a[1] = hi ? 1.0f : gy;
        const v8f d0 = wmma_dist(a, b0);
        const v8f d1 = wmma_dist(a, b1);
        const int rbase = t * 16 + (hi ? 8 : 0);
#pragma unroll
        for (int v = 0; v < 8; ++v) {
            const int   r  = rbase + v;
            const float e0 = d0[v];
            const bool  u0 = e0 < min0;
            min0 = u0 ? e0 : min0;
            idx0 = u0 ? r  : idx0;
            const float e1 = d1[v];
            const bool  u1 = e1 < min1;
            min1 = u1 ? e1 : min1;
            idx1 = u1 ? r  : idx1;
        }
    }
    // combine lane l with lane l^16 (rows 0..7 tile-half vs 8..15 tile-half),
    // first-occurrence tie-break like jnp.argmin.
    {
        float om = __shfl_xor(min0, 16, 32);
        int   oi = __shfl_xor(idx0, 16, 32);
        bool  u  = (om < min0) || (om == min0 && oi < idx0);
        min0 = u ? om : min0;  idx0 = u ? oi : idx0;
        om = __shfl_xor(min1, 16, 32);
        oi = __shfl_xor(idx1, 16, 32);
        u  = (om < min1) || (om == min1 && oi < idx1);
        min1 = u ? om : min1;  idx1 = u ? oi : idx1;
    }
    {
        // lanes 0..15 finalize tile c0's columns, lanes 16..31 tile c1's
        const int j   = (hi ? c1 : c0) * 16 + n;
        const int idx = hi ? idx1 : idx0;
        const float nx  = sIx[idx];
        const float ny  = sIy[idx];
        const float pdx = pred2[(b * NP + j) * 2 + 0];
        const float pdy = pred2[(b * NP + j) * 2 + 1];
        atomicAdd(&sAcc[0], fabsf(pdx - nx) + fabsf(pdy - ny));
    }

    // ================= stage 2: gt vertex -> nearest pred =================
    // transpose roles: rows (M) = ini_pred points, cols (N) = gt vertices,
    // so argmin-over-pred is again a row reduction.
    v2f g0, g1;
    {
        float gx = sGx[c0 * 16 + n], gy = sGy[c0 * 16 + n];
        g0[0] = hi ? 1.0f : -2.0f * gx;
        g0[1] = hi ? (gx * gx + gy * gy) : -2.0f * gy;
        gx = sGx[c1 * 16 + n]; gy = sGy[c1 * 16 + n];
        g1[0] = hi ? 1.0f : -2.0f * gx;
        g1[1] = hi ? (gx * gx + gy * gy) : -2.0f * gy;
    }
    min0 = 3.0e38f; min1 = 3.0e38f; idx0 = 0; idx1 = 0;
    for (int t = 0; t < NP / 16; ++t) {
        const float px = sPx[t * 16 + n];
        const float py = sPy[t * 16 + n];
        v2f a;
        a[0] = hi ? (px * px + py * py) : px;
        a[1] = hi ? 1.0f : py;
        const v8f d0 = wmma_dist(a, g0);
        const v8f d1 = wmma_dist(a, g1);
        const int rbase = t * 16 + (hi ? 8 : 0);
#pragma unroll
        for (int v = 0; v < 8; ++v) {
            const int   r  = rbase + v;
            const float e0 = d0[v];
            const bool  u0 = e0 < min0;
            min0 = u0 ? e0 : min0;
            idx0 = u0 ? r  : idx0;
            const float e1 = d1[v];
            const bool  u1 = e1 < min1;
            min1 = u1 ? e1 : min1;
            idx1 = u1 ? r  : idx1;
        }
    }
    {
        float om = __shfl_xor(min0, 16, 32);
        int   oi = __shfl_xor(idx0, 16, 32);
        bool  u  = (om < min0) || (om == min0 && oi < idx0);
        min0 = u ? om : min0;  idx0 = u ? oi : idx0;
        om = __shfl_xor(min1, 16, 32);
        oi = __shfl_xor(idx1, 16, 32);
        u  = (om < min1) || (om == min1 && oi < idx1);
        min1 = u ? om : min1;  idx1 = u ? oi : idx1;
    }
    {
        const int g   = (hi ? c1 : c0) * 16 + n;
        const int idx = hi ? idx1 : idx0;
        const float nx = pred2[(b * NP + idx) * 2 + 0];   // gather from pred_polys_
        const float ny = pred2[(b * NP + idx) * 2 + 1];
        const float gx = sGx[g];
        const float gy = sGy[g];
        const float mk = kmask[b * NG + g];
        atomicAdd(&sAcc[1], mk * (fabsf(nx - gx) + fabsf(ny - gy)));
        atomicAdd(&sAcc[2], 2.0f * mk);   // mask broadcast over 2 coords
    }
    __syncthreads();
    if (tid == 0) {
        atomicAdd(&ws[0], sAcc[0]);
        atomicAdd(&ws[1], sAcc[1]);
        atomicAdd(&ws[2], sAcc[2]);
    }
}

__global__ void dmloss_zero_ws(float* __restrict__ ws) {
    if (threadIdx.x < 3) ws[threadIdx.x] = 0.0f;
}

__global__ void dmloss_finalize(const float* __restrict__ ws,
                                float* __restrict__ out,
                                float inv_pred_count) {
    const float loss_pred2gt = ws[0] * inv_pred_count;           // mean L1
    const float loss_gt2pred = ws[1] / (ws[2] + 1.0f);           // masked sum / (sum mask + 1)
    out[0] = (loss_gt2pred + loss_pred2gt) * 0.5f;
}

extern "C" void kernel_launch(void* const* d_in, const int* in_sizes, int n_in,
                              void* d_out, int out_size, void* d_ws, size_t ws_size,
                              hipStream_t stream) {
    const float* ini_pred = (const float*)d_in[0];   // [B,NP,2]
    const float* pred2    = (const float*)d_in[1];   // [B,NP,2]
    const float* gtp      = (const float*)d_in[2];   // [B,NG,2]
    const float* kmask    = (const float*)d_in[3];   // [B,NG]
    float*       out      = (float*)d_out;
    float*       ws       = (float*)d_ws;

    const int B = in_sizes[0] / (NP * 2);

    dmloss_zero_ws<<<1, 32, 0, stream>>>(ws);
    dmloss_batch_kernel<<<B, 128, 0, stream>>>(ini_pred, pred2, gtp, kmask, ws);
    dmloss_finalize<<<1, 1, 0, stream>>>(ws, out, 1.0f / (float)(B * NP * 2));
}